// TJDLayer_19052474925143
// MI455X (gfx1250) — compile-verified
//
#include <hip/hip_runtime.h>
#include <hip/hip_bf16.h>

typedef __attribute__((ext_vector_type(2))) float v2f;
typedef __attribute__((ext_vector_type(8))) float v8f;

#define BB 8
#define TT 512
#define DD 1024
#define RR 4
#define VV 2048
#define CTOT (VV * RR * RR)   // 32768 columns of w_vocab
#define LROW (DD + 2)         // LDS row pad: keeps b64 alignment, breaks bank conflicts

// ---------------------------------------------------------------------------
// Kernel 1a/1b: xmean[b,d] = mean_T input_embs[b,t,d], two-stage so 256 blocks
// stream the 16.8 MB input (32 blocks would leave most WGPs idle).
// Stage 1: (b, d-slab, t-chunk) -> partial sums. Stage 2: fold 8 chunks.
// ---------------------------------------------------------------------------
__global__ void __launch_bounds__(256) tjd_mean1_kernel(
    const float* __restrict__ x, float* __restrict__ part) {
  const int b    = blockIdx.x >> 5;
  const int slab = (blockIdx.x >> 3) & 3;
  const int tc   = blockIdx.x & 7;
  const int d    = (slab << 8) + threadIdx.x;
  const float* p = x + ((size_t)(b * TT + tc * 64)) * DD + d;
  float acc = 0.0f;
#pragma unroll 8
  for (int t = 0; t < 64; ++t) acc += p[(size_t)t * DD];
  part[tc * (BB * DD) + b * DD + d] = acc;
}

__global__ void __launch_bounds__(256) tjd_mean2_kernel(
    const float* __restrict__ part, float* __restrict__ xmean) {
  const int b = blockIdx.x >> 2;
  const int d = ((blockIdx.x & 3) << 8) + threadIdx.x;
  float acc = 0.0f;
#pragma unroll
  for (int tc = 0; tc < 8; ++tc) acc += part[tc * (BB * DD) + b * DD + d];
  xmean[b * DD + d] = acc * (1.0f / (float)TT);
}

// ---------------------------------------------------------------------------
// Kernel 2: core[b,c] = sum_d xmean[b,d] * w_vocab[d,c] via V_WMMA_F32_16X16X4_F32.
// M padded 8->16 (rows 8..15 zero in LDS). One wave = one 16-column tile,
// 256 WMMA steps over K=1024. 512 blocks x 4 waves stream w_vocab (134 MB)
// exactly once -> HBM-bound (~6 us at 23.3 TB/s). Verified good codegen.
// ---------------------------------------------------------------------------
__global__ void __launch_bounds__(128) tjd_core_wmma_kernel(
    const float* __restrict__ xmean, const float* __restrict__ wv,
    float* __restrict__ core) {
  __shared__ float lA[16 * LROW];

  for (int idx = threadIdx.x; idx < 16 * LROW; idx += 128) {
    const int row = idx / LROW;
    const int col = idx - row * LROW;
    lA[idx] = (row < BB && col < DD) ? xmean[row * DD + col] : 0.0f;
  }
  __syncthreads();

  const int lane = threadIdx.x & 31;
  const int wave = threadIdx.x >> 5;
  const int n    = lane & 15;   // tile column / A-row (M)
  const int kh   = lane >> 4;   // K half
  const int col  = (blockIdx.x * 4 + wave) * 16 + n;

  const float* pA = &lA[n * LROW + 2 * kh];
  const float* pB = wv + (size_t)(2 * kh) * CTOT + col;

  v8f acc = {};
#pragma unroll 4
  for (int k0 = 0; k0 < DD; k0 += 4) {
    v2f a = *(const v2f*)pA;
    v2f bf;
    bf.x = pB[0];
    bf.y = pB[CTOT];
    acc = __builtin_amdgcn_wmma_f32_16x16x4_f32(
        false, a, false, bf, (short)0, acc, false, false);
    pA += 4;
    pB += (size_t)4 * CTOT;
  }

  if (lane < 16) {
#pragma unroll
    for (int r = 0; r < 8; ++r)
      core[(size_t)r * CTOT + col] = acc[r];
  }
}

// ---------------------------------------------------------------------------
// Kernel 3: everything downstream of core, in ONE 1024-thread workgroup using
// a 258 KB dynamic-LDS working set (CDNA5 WGP has 320 KB):
//   A) alpha/beta = xmean @ w_{alpha,beta}: 256 partial reducers + wave0 finish
//   B) gather + column-normalize all 4096 4x4 matrices in parallel -> LDS
//   C) associative tree product of the 512-matrix chain per batch: 9 levels
//      of parallel 4x4 matmuls (replaces a 512-deep serial latency chain)
//   D) v = P @ n(alpha); loss = mean_b(-log(n(beta)·v)) via wave0 shuffles
// ---------------------------------------------------------------------------
__global__ void __launch_bounds__(1024) tjd_chain_kernel(
    const float* __restrict__ xmean, const float* __restrict__ w_alpha,
    const float* __restrict__ w_beta, const int* __restrict__ labels,
    const float* __restrict__ core, float* __restrict__ out) {
  extern __shared__ float smem[];
  float* Mb    = smem;                       // 8*512 matrices, col-major 4x4 (65536 f)
  float* partA = Mb + BB * TT * 16;          // 256 f
  float* partB = partA + 256;                // 256 f
  float* an    = partB + 256;                // 32 f  (normalized alpha)
  float* bn    = an + 32;                    // 32 f  (normalized beta)

  const int tid = threadIdx.x;

  // --- Phase A (partials): thread = (b, r, chunk-of-128-d) ---
  if (tid < 256) {
    const int b = tid >> 5, rem = tid & 31, r = rem >> 3, ch = rem & 7;
    const float* xr = xmean + (b << 10) + (ch << 7);
    const float* wa = w_alpha + ((ch << 7) << 2) + r;
    const float* wb = w_beta + ((ch << 7) << 2) + r;
    float aa = 0.0f, ba = 0.0f;
#pragma unroll 4
    for (int d = 0; d < 128; ++d) {
      const float xv = xr[d];
      aa = fmaf(xv, wa[d << 2], aa);
      ba = fmaf(xv, wb[d << 2], ba);
    }
    partA[tid] = aa;  // tid == (b*4+r)*8 + ch
    partB[tid] = ba;
  }

  // --- Phase B: gather + column-normalize. task q = (b, t, column j) ---
  for (int q = tid; q < BB * TT * 4; q += 1024) {
    const int j = q & 3;
    const int t = (q >> 2) & (TT - 1);
    const int b = q >> 11;
    const int y = labels[(b << 9) + t];
    const float* cp = core + (size_t)b * CTOT + ((size_t)y << 2) + j;
    const float m0 = fabsf(cp[0]);                // i stride = V*R = 8192
    const float m1 = fabsf(cp[8192]);
    const float m2 = fabsf(cp[16384]);
    const float m3 = fabsf(cp[24576]);
    const float inv = 1.0f / (m0 + m1 + m2 + m3);
    float* dst = Mb + (((b << 9) + t) << 4) + (j << 2);
    dst[0] = m0 * inv; dst[1] = m1 * inv; dst[2] = m2 * inv; dst[3] = m3 * inv;
  }
  __syncthreads();

  // --- Phase A (finish): wave0 lanes, lane = b*4 + r ---
  if (tid < 32) {
    float aa = 0.0f, ba = 0.0f;
#pragma unroll
    for (int ch = 0; ch < 8; ++ch) {
      aa += partA[(tid << 3) + ch];
      ba += partB[(tid << 3) + ch];
    }
    aa = fabsf(aa); ba = fabsf(ba);
    float sa = aa + __shfl_xor(aa, 1, 32); sa += __shfl_xor(sa, 2, 32);
    float sb = ba + __shfl_xor(ba, 1, 32); sb += __shfl_xor(sb, 2, 32);
    an[tid] = aa / sa;
    bn[tid] = ba / sb;
  }

  // --- Phase C: tree product. Level l pairs (p*2^(l+1), +2^l); later matrix
  // left-multiplies earlier. Task = one column of one product; writes back
  // into the earlier matrix's slot (only its own column -> race-free). ---
  for (int l = 0; l < 9; ++l) {
    const int pairs = 256 >> l;               // per batch
    const int tasks = (BB * pairs) << 2;
    for (int q = tid; q < tasks; q += 1024) {
      const int j  = q & 3;
      const int pb = q >> 2;
      const int b  = pb >> (8 - l);
      const int p  = pb & (pairs - 1);
      float* A1       = Mb + (((b << 9) + (p << (l + 1))) << 4);  // earlier (right)
      const float* A2 = A1 + (16 << l);                            // later (left)
      const float a0 = A1[(j << 2) + 0];
      const float a1 = A1[(j << 2) + 1];
      const float a2 = A1[(j << 2) + 2];
      const float a3 = A1[(j << 2) + 3];
#pragma unroll
      for (int i = 0; i < 4; ++i) {
        A1[(j << 2) + i] = A2[i] * a0 + A2[4 + i] * a1 +
                           A2[8 + i] * a2 + A2[12 + i] * a3;
      }
    }
    __syncthreads();
  }

  // --- Phase D: v = P_b @ n(alpha_b); p_b = n(beta_b)·v; loss = mean(-log p) ---
  if (tid < 32) {
    const int b = tid >> 2, i = tid & 3;
    const float* P = Mb + (b << 13);          // b*512*16, product at slot 0
    const float* ab = an + (b << 2);
    float v = P[i] * ab[0] + P[4 + i] * ab[1] + P[8 + i] * ab[2] + P[12 + i] * ab[3];
    float p = bn[tid] * v;
    p += __shfl_xor(p, 1, 32);
    p += __shfl_xor(p, 2, 32);
    float contrib = (i == 0) ? (-logf(p) * (1.0f / (float)BB)) : 0.0f;
    contrib += __shfl_xor(contrib, 4, 32);
    contrib += __shfl_xor(contrib, 8, 32);
    contrib += __shfl_xor(contrib, 16, 32);
    if (tid == 0) out[0] = contrib;
  }
}

extern "C" void kernel_launch(void* const* d_in, const int* in_sizes, int n_in,
                              void* d_out, int out_size, void* d_ws, size_t ws_size,
                              hipStream_t stream) {
  const float* x       = (const float*)d_in[0];   // [8,512,1024] f32
  const int*   labels  = (const int*)d_in[1];     // [8,512] i32
  const float* w_alpha = (const float*)d_in[2];   // [1024,4] f32
  const float* w_beta  = (const float*)d_in[3];   // [1024,4] f32
  const float* w_vocab = (const float*)d_in[4];   // [1024,32768] f32
  float* out = (float*)d_out;

  float* xmean    = (float*)d_ws;                 // 8*1024 f32
  float* core     = xmean + BB * DD;              // 8*32768 f32 (1 MB)
  float* meanpart = core + (size_t)BB * CTOT;     // 8*8192 f32 (256 KB)

  tjd_mean1_kernel<<<256, 256, 0, stream>>>(x, meanpart);
  tjd_mean2_kernel<<<32, 256, 0, stream>>>(meanpart, xmean);
  tjd_core_wmma_kernel<<<CTOT / 64, 128, 0, stream>>>(xmean, w_vocab, core);

  const size_t smem_bytes = (size_t)(BB * TT * 16 + 256 + 256 + 32 + 32) * sizeof(float);
  tjd_chain_kernel<<<1, 1024, smem_bytes, stream>>>(xmean, w_alpha, w_beta,
                                                    labels, core, out);
}